// BootstrappedCE_72318659330453
// MI455X (gfx1250) — compile-verified
//
#include <hip/hip_runtime.h>

// ---------------------------------------------------------------------------
// Bootstrapped BCE (top-25% mean) for gfx1250 / MI455X.
//
// Exact top-k mean via 3-level (12+12+8 bit) radix select on the IEEE-754 bit
// pattern of the per-element BCE loss (monotone since loss >= 0).
//   Pass A: LDS-privatized 4096-bin histogram (bits 31:20) + full sum (WMMA);
//           if ws is large enough, also stores the 32-bit keys (64 MB) which
//           fit in MI455X's 192 MB L2 -> passes B/C run at L2 rate.
//           Inputs read non-temporally in that path (each byte read once).
//   Pass B: 4096-bin histogram (bits 19:8) of elements in selected L1 bin
//   Pass C: 256-bin histogram (bits 7:0) of elements in selected 24-bit bin
//           fused with sum of all elements strictly above that bin (WMMA)
//   Final : last 8 bits resolved analytically (equal keys == equal floats;
//           key bits reconstruct the exact loss value)
// Wave sum reductions use V_WMMA_F32_16X16X4_F32 with B = ones (row sums).
// ---------------------------------------------------------------------------

#define N_TOTAL (16u * 1024u * 1024u)   // 16 * 1 * 1024 * 1024
#define K_TOP   (N_TOTAL / 4u)          // TOP_P = 0.25
#define START_WARM 1000

typedef __attribute__((ext_vector_type(2))) float v2f;
typedef __attribute__((ext_vector_type(4))) float v4f;   // NT-load compatible
typedef __attribute__((ext_vector_type(8))) float v8f;

__device__ __forceinline__ float bce_val(float p, float t) {
    // matches torch F.binary_cross_entropy with logs clamped at -100
    float lp = fmaxf(__logf(p), -100.0f);
    float lq = fmaxf(__logf(1.0f - p), -100.0f);
    return -(t * lp + (1.0f - t) * lq);
}

// Sum 64 per-wave partials (2 per lane) with one V_WMMA_F32_16X16X4_F32.
// A (16x4) = the 64 partials; B (4x16) = ones; D[m][n] = rowsum(m).
// Each lane sums its 8 C VGPRs (half the M rows); lane^16 swap adds the rest.
__device__ __forceinline__ float wave_sum_wmma(float acc0, float acc1) {
    v2f a; a.x = acc0; a.y = acc1;
    v2f b; b.x = 1.0f; b.y = 1.0f;
    v8f c = {};
    c = __builtin_amdgcn_wmma_f32_16x16x4_f32(
        /*neg_a=*/false, a, /*neg_b=*/false, b,
        /*c_mod=*/(short)0, c, /*reuse_a=*/false, /*reuse_b=*/false);
    float h = c[0] + c[1] + c[2] + c[3] + c[4] + c[5] + c[6] + c[7];
    h += __shfl_xor(h, 16, 32);
    return h;
}

// ws layout (32-bit words):
//   [0    .. 4095]   h1: histogram of key[31:20]
//   [4096 .. 8191]   h2: histogram of key[19:8] within L1 bin
//   [8192 .. 8447]   h3: histogram of key[7:0]  within 24-bit bin
//   [8448 .. 8455]   ctrl: [0]=P1 [1]=C1 [2]=r1 [3]=P24 [4]=r2
//   [8456 .. 8457]   sums (float): [0]=full sum, [1]=sum above 24-bit bin
//   [16384 .. ]      optional key cache (N_TOTAL u32 = 64 MB, L2-resident)
#define WS_H2    4096u
#define WS_H3    8192u
#define WS_CTRL  8448u
#define WS_SUMS  8456u
#define WS_WORDS 8458u
#define WS_KEYS  16384u

__global__ void k_init(unsigned int* ws) {
    unsigned int i = blockIdx.x * blockDim.x + threadIdx.x;
    if (i < WS_WORDS) ws[i] = 0u;
}

// ---------------- Pass A: LDS-privatized L1 histogram + full sum -----------
// STORE_KEYS=1: also write 32-bit keys to ws (inputs read non-temporally,
// key stores left cacheable so the 64 MB array stays resident in 192 MB L2).
template <int STORE_KEYS>
__global__ __launch_bounds__(512)
void k_hist1(const float* __restrict__ p, const float* __restrict__ t,
             unsigned int* __restrict__ h1, float* __restrict__ sums,
             uint4* __restrict__ keys) {
    __shared__ unsigned int lh[4096];
    for (unsigned int j = threadIdx.x; j < 4096u; j += blockDim.x) lh[j] = 0u;
    __syncthreads();

    float acc0 = 0.0f, acc1 = 0.0f;
    unsigned int i = blockIdx.x * blockDim.x + threadIdx.x;
    unsigned int stride = gridDim.x * blockDim.x;
    for (unsigned int idx = i; idx < N_TOTAL / 4u; idx += stride) {
        v4f pv, tv;
        if (STORE_KEYS) {   // streamed exactly once -> TH=NT keeps L2 for keys
            pv = __builtin_nontemporal_load((const v4f*)(p + 4u * idx));
            tv = __builtin_nontemporal_load((const v4f*)(t + 4u * idx));
        } else {
            pv = *(const v4f*)(p + 4u * idx);
            tv = *(const v4f*)(t + 4u * idx);
        }
        float v0 = bce_val(pv.x, tv.x);
        float v1 = bce_val(pv.y, tv.y);
        float v2 = bce_val(pv.z, tv.z);
        float v3 = bce_val(pv.w, tv.w);
        unsigned int k0 = __float_as_uint(v0);
        unsigned int k1 = __float_as_uint(v1);
        unsigned int k2 = __float_as_uint(v2);
        unsigned int k3 = __float_as_uint(v3);
        if (STORE_KEYS) {
            uint4 kv; kv.x = k0; kv.y = k1; kv.z = k2; kv.w = k3;
            keys[idx] = kv;
        }
        atomicAdd(&lh[k0 >> 20], 1u);   // ds_add_u32
        atomicAdd(&lh[k1 >> 20], 1u);
        atomicAdd(&lh[k2 >> 20], 1u);
        atomicAdd(&lh[k3 >> 20], 1u);
        acc0 += v0 + v1;
        acc1 += v2 + v3;
    }

    // full-sum reduction: one v_wmma per wave, one atomic per wave
    float w = wave_sum_wmma(acc0, acc1);
    if ((threadIdx.x & 31u) == 0u) atomicAdd(&sums[0], w);

    __syncthreads();
    // aggregated flush: one global atomic per nonzero bin per block
    for (unsigned int j = threadIdx.x; j < 4096u; j += blockDim.x) {
        unsigned int c = lh[j];
        if (c != 0u) atomicAdd(&h1[j], c);
    }
}

// --------- generic select over a 4096-bin histogram (one 1024-thr block) ---
// mode 0: target = K_TOP;  writes ctrl[0]=P1, ctrl[1]=C1, ctrl[2]=r1
// mode 1: target = ctrl[2]; writes ctrl[3]=P24=(P1<<12)|B2, ctrl[4]=r2
__global__ __launch_bounds__(1024)
void k_select(const unsigned int* __restrict__ hist,
              unsigned int* __restrict__ ctrl, int mode) {
    __shared__ unsigned int chunk[1024];   // 4 bins per chunk
    __shared__ unsigned int super[32];     // 32 chunks per super
    int tid = threadIdx.x;
    unsigned int s = hist[tid * 4 + 0] + hist[tid * 4 + 1] +
                     hist[tid * 4 + 2] + hist[tid * 4 + 3];
    chunk[tid] = s;
    __syncthreads();
    if (tid < 32) {
        unsigned int ss = 0u;
        for (int j = 0; j < 32; ++j) ss += chunk[tid * 32 + j];
        super[tid] = ss;
    }
    __syncthreads();
    if (tid == 0) {
        unsigned int target = (mode == 0) ? K_TOP : ctrl[2];
        unsigned int running = 0u;
        int sstar = 0;
        for (int q = 31; q >= 0; --q) {
            if (running + super[q] >= target) { sstar = q; break; }
            running += super[q];
        }
        int cstar = sstar * 32;
        for (int c = sstar * 32 + 31; c >= sstar * 32; --c) {
            if (running + chunk[c] >= target) { cstar = c; break; }
            running += chunk[c];
        }
        unsigned int bstar = (unsigned int)(cstar * 4);
        for (int b = cstar * 4 + 3; b >= cstar * 4; --b) {
            unsigned int hb = hist[b];
            if (running + hb >= target) { bstar = (unsigned int)b; break; }
            running += hb;
        }
        if (mode == 0) {
            ctrl[0] = bstar;              // P1 (12-bit prefix)
            ctrl[1] = running;            // C1 = count strictly above bin
            ctrl[2] = target - running;   // r1
        } else {
            ctrl[3] = (ctrl[0] << 12) | bstar;  // P24 (24-bit prefix)
            ctrl[4] = target - running;         // r2
        }
    }
}

// ---------------- Pass B: L2 histogram within selected L1 bin --------------
__global__ __launch_bounds__(512)
void k_hist2_recompute(const float4* __restrict__ p, const float4* __restrict__ t,
                       const unsigned int* __restrict__ ctrl,
                       unsigned int* __restrict__ h2) {
    unsigned int P1 = ctrl[0];
    unsigned int i = blockIdx.x * blockDim.x + threadIdx.x;
    unsigned int stride = gridDim.x * blockDim.x;
    for (unsigned int idx = i; idx < N_TOTAL / 4u; idx += stride) {
        float4 pv = p[idx];
        float4 tv = t[idx];
        unsigned int k0 = __float_as_uint(bce_val(pv.x, tv.x));
        unsigned int k1 = __float_as_uint(bce_val(pv.y, tv.y));
        unsigned int k2 = __float_as_uint(bce_val(pv.z, tv.z));
        unsigned int k3 = __float_as_uint(bce_val(pv.w, tv.w));
        if ((k0 >> 20) == P1) atomicAdd(&h2[(k0 >> 8) & 0xFFFu], 1u);
        if ((k1 >> 20) == P1) atomicAdd(&h2[(k1 >> 8) & 0xFFFu], 1u);
        if ((k2 >> 20) == P1) atomicAdd(&h2[(k2 >> 8) & 0xFFFu], 1u);
        if ((k3 >> 20) == P1) atomicAdd(&h2[(k3 >> 8) & 0xFFFu], 1u);
    }
}

__global__ __launch_bounds__(512)
void k_hist2_keys(const uint4* __restrict__ keys,
                  const unsigned int* __restrict__ ctrl,
                  unsigned int* __restrict__ h2) {
    unsigned int P1 = ctrl[0];
    unsigned int i = blockIdx.x * blockDim.x + threadIdx.x;
    unsigned int stride = gridDim.x * blockDim.x;
    for (unsigned int idx = i; idx < N_TOTAL / 4u; idx += stride) {
        uint4 kv = keys[idx];   // L2-resident
        if ((kv.x >> 20) == P1) atomicAdd(&h2[(kv.x >> 8) & 0xFFFu], 1u);
        if ((kv.y >> 20) == P1) atomicAdd(&h2[(kv.y >> 8) & 0xFFFu], 1u);
        if ((kv.z >> 20) == P1) atomicAdd(&h2[(kv.z >> 8) & 0xFFFu], 1u);
        if ((kv.w >> 20) == P1) atomicAdd(&h2[(kv.w >> 8) & 0xFFFu], 1u);
    }
}

// --- Pass C (fused): L3 histogram within 24-bit bin + sum strictly above ---
__global__ __launch_bounds__(512)
void k_pass3_recompute(const float4* __restrict__ p, const float4* __restrict__ t,
                       const unsigned int* __restrict__ ctrl,
                       unsigned int* __restrict__ h3, float* __restrict__ sums) {
    unsigned int P24 = ctrl[3];
    float acc0 = 0.0f, acc1 = 0.0f;
    unsigned int i = blockIdx.x * blockDim.x + threadIdx.x;
    unsigned int stride = gridDim.x * blockDim.x;
    for (unsigned int idx = i; idx < N_TOTAL / 4u; idx += stride) {
        float4 pv = p[idx];
        float4 tv = t[idx];
        float v0 = bce_val(pv.x, tv.x);
        float v1 = bce_val(pv.y, tv.y);
        float v2 = bce_val(pv.z, tv.z);
        float v3 = bce_val(pv.w, tv.w);
        unsigned int k0 = __float_as_uint(v0) >> 8;
        unsigned int k1 = __float_as_uint(v1) >> 8;
        unsigned int k2 = __float_as_uint(v2) >> 8;
        unsigned int k3 = __float_as_uint(v3) >> 8;
        acc0 += (k0 > P24 ? v0 : 0.0f);
        acc0 += (k1 > P24 ? v1 : 0.0f);
        acc1 += (k2 > P24 ? v2 : 0.0f);
        acc1 += (k3 > P24 ? v3 : 0.0f);
        if (k0 == P24) atomicAdd(&h3[__float_as_uint(v0) & 0xFFu], 1u);
        if (k1 == P24) atomicAdd(&h3[__float_as_uint(v1) & 0xFFu], 1u);
        if (k2 == P24) atomicAdd(&h3[__float_as_uint(v2) & 0xFFu], 1u);
        if (k3 == P24) atomicAdd(&h3[__float_as_uint(v3) & 0xFFu], 1u);
    }
    float w = wave_sum_wmma(acc0, acc1);
    if ((threadIdx.x & 31u) == 0u) atomicAdd(&sums[1], w);
}

__global__ __launch_bounds__(512)
void k_pass3_keys(const uint4* __restrict__ keys,
                  const unsigned int* __restrict__ ctrl,
                  unsigned int* __restrict__ h3, float* __restrict__ sums) {
    unsigned int P24 = ctrl[3];
    float acc0 = 0.0f, acc1 = 0.0f;
    unsigned int i = blockIdx.x * blockDim.x + threadIdx.x;
    unsigned int stride = gridDim.x * blockDim.x;
    for (unsigned int idx = i; idx < N_TOTAL / 4u; idx += stride) {
        uint4 kv = keys[idx];   // key bits == exact loss value bits
        acc0 += ((kv.x >> 8) > P24 ? __uint_as_float(kv.x) : 0.0f);
        acc0 += ((kv.y >> 8) > P24 ? __uint_as_float(kv.y) : 0.0f);
        acc1 += ((kv.z >> 8) > P24 ? __uint_as_float(kv.z) : 0.0f);
        acc1 += ((kv.w >> 8) > P24 ? __uint_as_float(kv.w) : 0.0f);
        if ((kv.x >> 8) == P24) atomicAdd(&h3[kv.x & 0xFFu], 1u);
        if ((kv.y >> 8) == P24) atomicAdd(&h3[kv.y & 0xFFu], 1u);
        if ((kv.z >> 8) == P24) atomicAdd(&h3[kv.z & 0xFFu], 1u);
        if ((kv.w >> 8) == P24) atomicAdd(&h3[kv.w & 0xFFu], 1u);
    }
    float w = wave_sum_wmma(acc0, acc1);
    if ((threadIdx.x & 31u) == 0u) atomicAdd(&sums[1], w);
}

// --------- Final: resolve last 8 bits analytically, emit both outputs ------
__global__ __launch_bounds__(256)
void k_final(const int* __restrict__ itp,
             const unsigned int* __restrict__ ctrl,
             const unsigned int* __restrict__ h3,
             const float* __restrict__ sums,
             float* __restrict__ out) {
    __shared__ unsigned int lh[256];
    lh[threadIdx.x] = h3[threadIdx.x];
    __syncthreads();
    if (threadIdx.x == 0) {
        int it = itp[0];
        if (it < START_WARM) {
            out[0] = sums[0] / (float)N_TOTAL;
            out[1] = 1.0f;
        } else {
            unsigned int P24 = ctrl[3];
            unsigned int r2  = ctrl[4];
            unsigned int running = 0u;
            int bstar = 0;
            float tail = 0.0f;   // exact: equal 32-bit keys are equal floats
            for (int b = 255; b >= 0; --b) {
                unsigned int hb = lh[b];
                if (running + hb >= r2) { bstar = b; break; }
                running += hb;
                tail += (float)hb * __uint_as_float((P24 << 8) | (unsigned int)b);
            }
            unsigned int need = r2 - running;  // # threshold-valued elements
            float thr = __uint_as_float((P24 << 8) | (unsigned int)bstar);
            float s = sums[1] + tail + (float)need * thr;
            out[0] = s / (float)K_TOP;
            out[1] = 0.25f;
        }
    }
}

extern "C" void kernel_launch(void* const* d_in, const int* in_sizes, int n_in,
                              void* d_out, int out_size, void* d_ws, size_t ws_size,
                              hipStream_t stream) {
    (void)in_sizes; (void)n_in; (void)out_size;
    const float* pf = (const float*)d_in[0];
    const float* tf = (const float*)d_in[1];
    const float4* p = (const float4*)d_in[0];
    const float4* t = (const float4*)d_in[1];
    const int* itp  = (const int*)d_in[2];
    unsigned int* ws = (unsigned int*)d_ws;
    unsigned int* h1   = ws;
    unsigned int* h2   = ws + WS_H2;
    unsigned int* h3   = ws + WS_H3;
    unsigned int* ctrl = ws + WS_CTRL;
    float* sums        = (float*)(ws + WS_SUMS);
    uint4* keys        = (uint4*)(ws + WS_KEYS);
    float* out = (float*)d_out;

    const int threads = 512;   // 16 wave32 per block
    const int blocks  = 1024;  // grid-stride over 4M float4s

    // Deterministic launch-time choice: key cache needs 64 MB + 64 KB of ws.
    const bool use_keys =
        ws_size >= (size_t)WS_KEYS * 4u + (size_t)N_TOTAL * 4u;

    k_init<<<(WS_WORDS + 255) / 256, 256, 0, stream>>>(ws);
    if (use_keys) {
        k_hist1<1><<<blocks, threads, 0, stream>>>(pf, tf, h1, sums, keys);
        k_select<<<1, 1024, 0, stream>>>(h1, ctrl, 0);
        k_hist2_keys<<<blocks, threads, 0, stream>>>(keys, ctrl, h2);
        k_select<<<1, 1024, 0, stream>>>(h2, ctrl, 1);
        k_pass3_keys<<<blocks, threads, 0, stream>>>(keys, ctrl, h3, sums);
    } else {
        k_hist1<0><<<blocks, threads, 0, stream>>>(pf, tf, h1, sums, keys);
        k_select<<<1, 1024, 0, stream>>>(h1, ctrl, 0);
        k_hist2_recompute<<<blocks, threads, 0, stream>>>(p, t, ctrl, h2);
        k_select<<<1, 1024, 0, stream>>>(h2, ctrl, 1);
        k_pass3_recompute<<<blocks, threads, 0, stream>>>(p, t, ctrl, h3, sums);
    }
    k_final<<<1, 256, 0, stream>>>(itp, ctrl, h3, sums, out);
}